// SheafWindowLayer_85779086835859
// MI455X (gfx1250) — compile-verified
//
#include <hip/hip_runtime.h>

// SheafWindowLayer on MI455X (gfx1250, wave32).
// fp32 end-to-end using V_WMMA_F32_16X16X4_F32 for both GEMMs.

typedef __attribute__((ext_vector_type(2))) float v2f;
typedef __attribute__((ext_vector_type(8))) float v8f;

#define BSZ     2
#define S_      512
#define HID_    768
#define MH_     128
#define M_TOT   (BSZ * S_)      // 1024 rows total
#define HALFWIN 256             // WIN/2

// ---------------------------------------------------------------------------
// Kernel 1: x = hidden_states + pos_emb[:S]  (pos broadcast over batch)
// ---------------------------------------------------------------------------
__global__ void add_pos_kernel(const float* __restrict__ hs,
                               const float* __restrict__ pos,
                               float* __restrict__ x) {
    int idx = blockIdx.x * blockDim.x + threadIdx.x;   // < M_TOT*HID_
    int sd  = idx % (S_ * HID_);                       // (s, d) within one batch
    x[idx] = hs[idx] + pos[sd];
}

// ---------------------------------------------------------------------------
// Kernel 2: A = x @ W1a + b1  (z==0)   /   Bm = x @ W1b  (z==1)
// One wave32 per 16x16 output tile. K staged through LDS in 16x64 chunks.
// WMMA fp32 layouts (ISA 7.12.2):
//   A 16x4 : lane l(<16) v0=A[M=l,K=0], v1=A[M=l,K=1]; lane l+16 -> K=2,3
//   B 4x16 : lane l(<16) v0=B[K=0,N=l], v1=B[K=1,N=l]; lane l+16 -> K=2,3
//   C 16x16: VGPR r holds M=r (lanes 0-15) and M=r+8 (lanes 16-31), N=lane&15
// ---------------------------------------------------------------------------
__global__ void gemm1_wmma(const float* __restrict__ X,    // M_TOT x HID_
                           const float* __restrict__ W1,   // 2*HID_ x MH_
                           const float* __restrict__ b1,   // MH_
                           float* __restrict__ Aout,       // M_TOT x MH_ (a+b1)
                           float* __restrict__ Bout) {     // M_TOT x MH_ (b)
    __shared__ float sX[16 * 68];                 // 16x64 tile, pad->68 floats
    const int lane  = threadIdx.x;                // 0..31
    const int half  = lane >> 4;                  // 0 / 1
    const int l     = lane & 15;
    const int m0    = blockIdx.x * 16;
    const int n0    = blockIdx.y * 16;
    const int which = blockIdx.z;                 // 0: W1a(+b1), 1: W1b
    const float* W  = W1 + which * (HID_ * MH_);

    v8f c = {};
    for (int k0 = 0; k0 < HID_; k0 += 64) {
        __syncthreads();
        // stage 16x64 of X with float4 loads (coalesced, 16B-aligned)
        for (int t = lane; t < 256; t += 32) {
            int row = t >> 4;
            int col = (t & 15) << 2;
            const float4 v = *(const float4*)(X + (m0 + row) * HID_ + k0 + col);
            *(float4*)(&sX[row * 68 + col]) = v;
        }
        __syncthreads();
        #pragma unroll
        for (int kk = 0; kk < 64; kk += 4) {
            v2f a;
            a.x = sX[l * 68 + kk + 2 * half + 0];
            a.y = sX[l * 68 + kk + 2 * half + 1];
            const int krow = k0 + kk + 2 * half;
            v2f b;
            b.x = W[(krow + 0) * MH_ + n0 + l];
            b.y = W[(krow + 1) * MH_ + n0 + l];
            c = __builtin_amdgcn_wmma_f32_16x16x4_f32(
                    /*neg_a=*/false, a, /*neg_b=*/false, b,
                    /*c_mod=*/(short)0, c, /*reuse_a=*/false, /*reuse_b=*/false);
        }
    }

    const float bias = (which == 0) ? b1[n0 + l] : 0.0f;
    float* Out = (which == 0) ? Aout : Bout;
    #pragma unroll
    for (int r = 0; r < 8; ++r) {
        Out[(m0 + r + 8 * half) * MH_ + n0 + l] = c[r] + bias;
    }
}

// ---------------------------------------------------------------------------
// Kernel 3: H[row,m] = mean_{j in window} relu(A[row,m] + Bm[bi,j,m])
// A already contains +b1. Window j in [max(0,i-256), min(S-1,i+255)],
// identical to the reference's clip+valid-mask formulation.
// ---------------------------------------------------------------------------
__global__ void window_relu_mean(const float* __restrict__ A,   // M_TOT x MH_
                                 const float* __restrict__ Bm,  // M_TOT x MH_
                                 float* __restrict__ H) {       // M_TOT x MH_
    const int row = blockIdx.x;          // bi*S + i
    const int m   = threadIdx.x;         // 0..127
    const int bi  = row / S_;
    const int i   = row - bi * S_;
    const int jlo = (i - HALFWIN) > 0 ? (i - HALFWIN) : 0;
    const int jhi = (i + HALFWIN - 1) < (S_ - 1) ? (i + HALFWIN - 1) : (S_ - 1);

    const float av = A[row * MH_ + m];
    const float* bp = Bm + (bi * S_) * MH_ + m;
    float acc = 0.0f;
    for (int j = jlo; j <= jhi; ++j) {
        acc += fmaxf(av + bp[j * MH_], 0.0f);    // coalesced 512B per j, L2-hot
    }
    H[row * MH_ + m] = acc / (float)(jhi - jlo + 1);
}

// ---------------------------------------------------------------------------
// Kernel 4: out = x + alpha * ((H @ W2 + b2) - x)
// One wave32 per 16x16 tile; full K=128 staged in LDS (16x132 padded).
// ---------------------------------------------------------------------------
__global__ void gemm2_wmma(const float* __restrict__ H,    // M_TOT x MH_
                           const float* __restrict__ W2,   // MH_ x HID_
                           const float* __restrict__ b2,   // HID_
                           const float* __restrict__ X,    // M_TOT x HID_
                           const float* __restrict__ alpha_p,
                           float* __restrict__ Out) {      // M_TOT x HID_
    __shared__ float sH[16 * 132];                // 16x128 tile, pad->132
    const int lane = threadIdx.x;
    const int half = lane >> 4;
    const int l    = lane & 15;
    const int m0   = blockIdx.x * 16;
    const int n0   = blockIdx.y * 16;

    for (int t = lane; t < 512; t += 32) {        // 16 rows x 32 float4
        int row = t >> 5;
        int col = (t & 31) << 2;
        const float4 v = *(const float4*)(H + (m0 + row) * MH_ + col);
        *(float4*)(&sH[row * 132 + col]) = v;
    }
    __syncthreads();

    v8f c = {};
    #pragma unroll
    for (int kk = 0; kk < MH_; kk += 4) {
        v2f a;
        a.x = sH[l * 132 + kk + 2 * half + 0];
        a.y = sH[l * 132 + kk + 2 * half + 1];
        const int krow = kk + 2 * half;
        v2f b;
        b.x = W2[(krow + 0) * HID_ + n0 + l];
        b.y = W2[(krow + 1) * HID_ + n0 + l];
        c = __builtin_amdgcn_wmma_f32_16x16x4_f32(
                false, a, false, b, (short)0, c, false, false);
    }

    const float alpha = *alpha_p;
    const float bias  = b2[n0 + l];
    #pragma unroll
    for (int r = 0; r < 8; ++r) {
        const int m   = m0 + r + 8 * half;
        const float xv = X[m * HID_ + n0 + l];
        Out[m * HID_ + n0 + l] = xv + alpha * ((c[r] + bias) - xv);
    }
}

// ---------------------------------------------------------------------------
extern "C" void kernel_launch(void* const* d_in, const int* in_sizes, int n_in,
                              void* d_out, int out_size, void* d_ws, size_t ws_size,
                              hipStream_t stream) {
    (void)in_sizes; (void)n_in; (void)out_size; (void)ws_size;

    const float* hs    = (const float*)d_in[0];  // (2,512,768)
    const float* pos   = (const float*)d_in[1];  // (512,768)
    const float* W1    = (const float*)d_in[2];  // (1536,128)
    const float* b1    = (const float*)d_in[3];  // (128,)
    const float* W2    = (const float*)d_in[4];  // (128,768)
    const float* b2    = (const float*)d_in[5];  // (768,)
    const float* alpha = (const float*)d_in[6];  // scalar

    float* out = (float*)d_out;                  // (2,512,768)
    float* ws  = (float*)d_ws;
    float* X   = ws;                             // 786432 floats
    float* A   = X  + (size_t)M_TOT * HID_;      // 131072 floats (a + b1)
    float* Bm  = A  + (size_t)M_TOT * MH_;       // 131072 floats (b)
    float* Hh  = Bm + (size_t)M_TOT * MH_;       // 131072 floats (windowed mean)

    add_pos_kernel<<<dim3((M_TOT * HID_) / 256), dim3(256), 0, stream>>>(hs, pos, X);

    gemm1_wmma<<<dim3(M_TOT / 16, MH_ / 16, 2), dim3(32), 0, stream>>>(
        X, W1, b1, A, Bm);

    window_relu_mean<<<dim3(M_TOT), dim3(MH_), 0, stream>>>(A, Bm, Hh);

    gemm2_wmma<<<dim3(M_TOT / 16, HID_ / 16), dim3(32), 0, stream>>>(
        Hh, W2, b2, X, alpha, out);
}